// GraphEmbedding_74612171866521
// MI455X (gfx1250) — compile-verified
//
#include <hip/hip_runtime.h>
#include <math.h>

typedef float v2f __attribute__((ext_vector_type(2)));
typedef float v8f __attribute__((ext_vector_type(8)));

#define N_NODES_T 100000

// ---------------------------------------------------------------- utilities

__global__ void fill_zero_kernel(float* __restrict__ p, int n) {
    int t = blockIdx.x * blockDim.x + threadIdx.x;
    if (t < n) p[t] = 0.0f;
}

__global__ void degree_kernel(const int* __restrict__ dst, float* __restrict__ deg, int E) {
    int t = blockIdx.x * blockDim.x + threadIdx.x;
    if (t < E) atomicAdd(&deg[dst[t]], 1.0f);
}

__global__ void rsqrt_kernel(float* __restrict__ d, int n) {
    int t = blockIdx.x * blockDim.x + threadIdx.x;
    if (t < n) d[t] = rsqrtf(d[t] + 1.0f);   // +1 for self-loop
}

// ------------------------------------------------- fp32 WMMA GEMM  Y = X @ W
// X: [M,K] row-major, W: [K,NOUT] row-major, Y: [M,NOUT].
// One wave computes one 16x16 tile of Y using V_WMMA_F32_16X16X4_F32.
// Layouts (ISA 7.12.2, 32-bit):
//   A(16x4):  lane = m + 16*(k>>1), vgpr = k&1  -> contiguous float2 per lane
//   B(4x16):  lane = n + 16*(k>>1), vgpr = k&1
//   C/D:      vgpr r, lane l -> row = r + 8*(l>=16), col = l&15
template <int K, int NOUT>
__global__ void gemm_wmma_kernel(const float* __restrict__ X,
                                 const float* __restrict__ W,
                                 float* __restrict__ Y, int M) {
    const int lane = threadIdx.x & 31;
    const int wid  = (blockIdx.x * blockDim.x + threadIdx.x) >> 5;
    const int tilesN = NOUT / 16;
    const int tilesM = M / 16;
    if (wid >= tilesM * tilesN) return;          // wave-uniform exit
    const int tm = wid / tilesN;
    const int tn = wid - tm * tilesN;

    const int g    = lane >> 4;                  // 0 or 1: K-pair group
    const int l15  = lane & 15;
    const int rowA = tm * 16 + l15;              // A row owned by this lane
    const int colB = tn * 16 + l15;              // B col owned by this lane

    v8f acc = {};
    for (int k = 0; k < K; k += 4) {
        const int ka = k + 2 * g;
        v2f a;
        a.x = X[rowA * K + ka];
        a.y = X[rowA * K + ka + 1];
        v2f b;
        b.x = W[ka * NOUT + colB];
        b.y = W[(ka + 1) * NOUT + colB];
        acc = __builtin_amdgcn_wmma_f32_16x16x4_f32(
            /*neg_a=*/false, a, /*neg_b=*/false, b,
            /*c_mod=*/(short)0, acc, /*reuse_a=*/false, /*reuse_b=*/false);
    }

    const int row0 = tm * 16 + 8 * g;
    const int col  = tn * 16 + l15;
#pragma unroll
    for (int r = 0; r < 8; ++r)
        Y[(row0 + r) * NOUT + col] = acc[r];
}

// --------------------------------------------- edge scatter (segment_sum)
// agg[dst] += h[src] * dis[src]*dis[dst].  One wave per edge; lane covers
// VEC contiguous features.  F == 32*VEC.
template <int F, int VEC>
__global__ void scatter_kernel(const int* __restrict__ src, const int* __restrict__ dst,
                               const float* __restrict__ h, const float* __restrict__ dis,
                               float* __restrict__ agg, int E) {
    int t = blockIdx.x * blockDim.x + threadIdx.x;
    int e = t >> 5;
    int lane = t & 31;
    if (e >= E) return;
    int s = src[e], d = dst[e];
    float norm = dis[s] * dis[d];
    const float* hp = h + (size_t)s * F + lane * VEC;
    float* ap = agg + (size_t)d * F + lane * VEC;
#pragma unroll
    for (int i = 0; i < VEC; ++i)
        atomicAdd(ap + i, hp[i] * norm);
}

// out = (agg + hlin*dis^2 + bias) [relu]
template <bool RELU>
__global__ void finalize_kernel(const float* __restrict__ hlin, const float* __restrict__ agg,
                                const float* __restrict__ dis, const float* __restrict__ bias,
                                float* __restrict__ out, int N, int F) {
    int t = blockIdx.x * blockDim.x + threadIdx.x;
    if (t >= N * F) return;
    int i = t / F;
    int f = t - i * F;
    float di = dis[i];
    float v = agg[t] + hlin[t] * di * di + bias[f];
    if (RELU) v = fmaxf(v, 0.0f);
    out[t] = v;
}

// ------------------------------------------------------- attention pooling
// column (optionally score-weighted) sum of h3 [N,64] -> out64[64]
template <bool WEIGHTED>
__global__ void colsum_kernel(const float* __restrict__ h, const float* __restrict__ w,
                              float* __restrict__ out64, int N) {
    const int CHUNK = 256;
    int f = threadIdx.x;           // blockDim.x == 64
    int n0 = blockIdx.x * CHUNK;
    int nEnd = min(n0 + CHUNK, N);
    float s = 0.0f;
    for (int n = n0; n < nEnd; ++n) {
        float v = h[(size_t)n * 64 + f];
        if (WEIGHTED) v *= w[n];
        s += v;
    }
    atomicAdd(&out64[f], s);
}

// ctx[j] = tanh( (1/N) * sum_k colsum[k] * W_att[k*64+j] ), single block of 64
__global__ void ctx_kernel(const float* __restrict__ colsum, const float* __restrict__ W_att,
                           float* __restrict__ ctx, int N) {
    int j = threadIdx.x;
    float invN = 1.0f / (float)N;
    float s = 0.0f;
    for (int k = 0; k < 64; ++k)
        s += colsum[k] * W_att[k * 64 + j];
    ctx[j] = tanhf(s * invN);
}

// scores[n] = sigmoid( dot(h3[n,:], ctx) ), one wave per node, float2 per lane
__global__ void scores_kernel(const float* __restrict__ h3, const float* __restrict__ ctx,
                              float* __restrict__ scores, int N) {
    int t = blockIdx.x * blockDim.x + threadIdx.x;
    int n = t >> 5;
    int lane = t & 31;
    if (n >= N) return;
    const float* hp = h3 + (size_t)n * 64 + lane * 2;
    float p = hp[0] * ctx[lane * 2] + hp[1] * ctx[lane * 2 + 1];
#pragma unroll
    for (int m = 16; m >= 1; m >>= 1)
        p += __shfl_xor(p, m, 32);
    if (lane == 0) scores[n] = 1.0f / (1.0f + expf(-p));
}

// ---------------------------------------------------------------- launcher

static inline int ceil_div(long long a, long long b) { return (int)((a + b - 1) / b); }

extern "C" void kernel_launch(void* const* d_in, const int* in_sizes, int n_in,
                              void* d_out, int out_size, void* d_ws, size_t ws_size,
                              hipStream_t stream) {
    (void)n_in; (void)out_size; (void)ws_size;

    const int* edge = (const int*)d_in[0];       // [2,E] int32
    const float* x  = (const float*)d_in[1];     // [N,128]
    const float* W1 = (const float*)d_in[2];
    const float* b1 = (const float*)d_in[3];
    const float* W2 = (const float*)d_in[4];
    const float* b2 = (const float*)d_in[5];
    const float* W3 = (const float*)d_in[6];
    const float* b3 = (const float*)d_in[7];
    const float* Wa = (const float*)d_in[8];

    const int E = in_sizes[0] / 2;
    const int N = in_sizes[1] / 128;
    const int* src = edge;
    const int* dst = edge + E;

    // workspace layout (floats)
    float* ws = (float*)d_ws;
    float* bufA   = ws;                          // N*128
    float* bufB   = bufA + (size_t)N * 128;      // N*128
    float* dis    = bufB + (size_t)N * 128;      // N
    float* scores = dis + N;                     // N
    float* colsum = scores + N;                  // 64
    float* ctx    = colsum + 64;                 // 64

    float* h3_out = (float*)d_out;               // [N,64]
    float* pooled = h3_out + (size_t)N * 64;     // [64]

    const int B = 256;

    // ---- degrees -> dis = rsqrt(deg+1)
    fill_zero_kernel<<<ceil_div(N, B), B, 0, stream>>>(dis, N);
    degree_kernel<<<ceil_div(E, B), B, 0, stream>>>(dst, dis, E);
    rsqrt_kernel<<<ceil_div(N, B), B, 0, stream>>>(dis, N);

    const int wavesPerBlock = B / 32;

    // ---- layer 1: bufA = x@W1 ; bufB = agg ; bufB = relu(...)
    {
        int tiles = (N / 16) * (128 / 16);
        gemm_wmma_kernel<128, 128><<<ceil_div(tiles, wavesPerBlock), B, 0, stream>>>(x, W1, bufA, N);
        fill_zero_kernel<<<ceil_div((long long)N * 128, B), B, 0, stream>>>(bufB, N * 128);
        scatter_kernel<128, 4><<<ceil_div((long long)E * 32, B), B, 0, stream>>>(src, dst, bufA, dis, bufB, E);
        finalize_kernel<true><<<ceil_div((long long)N * 128, B), B, 0, stream>>>(bufA, bufB, dis, b1, bufB, N, 128);
    }
    // ---- layer 2: bufA = h1@W2 ; bufB = agg ; bufB = relu(...)
    {
        int tiles = (N / 16) * (128 / 16);
        gemm_wmma_kernel<128, 128><<<ceil_div(tiles, wavesPerBlock), B, 0, stream>>>(bufB, W2, bufA, N);
        fill_zero_kernel<<<ceil_div((long long)N * 128, B), B, 0, stream>>>(bufB, N * 128);
        scatter_kernel<128, 4><<<ceil_div((long long)E * 32, B), B, 0, stream>>>(src, dst, bufA, dis, bufB, E);
        finalize_kernel<true><<<ceil_div((long long)N * 128, B), B, 0, stream>>>(bufA, bufB, dis, b2, bufB, N, 128);
    }
    // ---- layer 3 (F3=64, no relu): h3_lin in bufA[0:N*64], agg in bufA[N*64:]
    {
        float* h3lin = bufA;
        float* agg3  = bufA + (size_t)N * 64;
        int tiles = (N / 16) * (64 / 16);
        gemm_wmma_kernel<128, 64><<<ceil_div(tiles, wavesPerBlock), B, 0, stream>>>(bufB, W3, h3lin, N);
        fill_zero_kernel<<<ceil_div((long long)N * 64, B), B, 0, stream>>>(agg3, N * 64);
        scatter_kernel<64, 2><<<ceil_div((long long)E * 32, B), B, 0, stream>>>(src, dst, h3lin, dis, agg3, E);
        finalize_kernel<false><<<ceil_div((long long)N * 64, B), B, 0, stream>>>(h3lin, agg3, dis, b3, h3_out, N, 64);
    }

    // ---- attention pooling
    fill_zero_kernel<<<1, 64, 0, stream>>>(colsum, 64);
    colsum_kernel<false><<<ceil_div(N, 256), 64, 0, stream>>>(h3_out, nullptr, colsum, N);
    ctx_kernel<<<1, 64, 0, stream>>>(colsum, Wa, ctx, N);
    scores_kernel<<<ceil_div((long long)N * 32, B), B, 0, stream>>>(h3_out, ctx, scores, N);
    fill_zero_kernel<<<1, 64, 0, stream>>>(pooled, 64);
    colsum_kernel<true><<<ceil_div(N, 256), 64, 0, stream>>>(h3_out, scores, pooled, N);
}